// NolinerSEM_5420248727802
// MI455X (gfx1250) — compile-verified
//
#include <hip/hip_runtime.h>

// CDNA5 / gfx1250. wave32. f32 WMMA: V_WMMA_F32_16X16X4_F32.
typedef float v2f __attribute__((ext_vector_type(2)));
typedef float v8f __attribute__((ext_vector_type(8)));

#define N_MODELS 90

__device__ __forceinline__ float leaky(float v) {
    return v >= 0.0f ? v : 0.33f * v;
}

// One wave (32 threads) per block; each block produces the constants for 16
// experts. The convs never see x (stride-2 over the padded width-1 axis only
// samples padding), so each expert's output is a weight-only constant.
__global__ void __launch_bounds__(32)
expert_const_kernel(const float* __restrict__ b1,
                    const float* __restrict__ W2,
                    const float* __restrict__ b2,
                    const float* __restrict__ W3,
                    const float* __restrict__ b3,
                    const float* __restrict__ Wfc,
                    const float* __restrict__ bfc,
                    float* __restrict__ consts) {
    __shared__ float v_int_s[32], v_top_s[32], v_bot_s[32];
    __shared__ float mean_s[16][64];   // pooled features, 16 experts
    __shared__ float dmat[16 * 16];    // WMMA D matrix spill

    const int lane = threadIdx.x;          // 0..31
    const int i0   = blockIdx.x * 16;      // first expert of this block

    for (int e = 0; e < 16; ++e) {
        int i  = i0 + e;
        int ic = (i < N_MODELS) ? i : (N_MODELS - 1);  // clamp (masked later)

        // conv1 output == leaky(b1) everywhere (x never sampled).
        float a1[16];
        #pragma unroll
        for (int c1 = 0; c1 < 16; ++c1)
            a1[c1] = leaky(b1[ic * 16 + c1]);

        // conv2 tap sums; lane == output channel c2. Three H classes:
        // h=0 (no kh=0 tap), interior, h=22 (no kh=2 tap). W2: [90,32,16,3,1].
        {
            const int c2 = lane;
            const float* w = W2 + ((size_t)ic * 32 + c2) * 16 * 3;
            float s0 = 0.f, s1 = 0.f, s2 = 0.f;
            #pragma unroll
            for (int c1 = 0; c1 < 16; ++c1) {
                float a = a1[c1];
                s0 = fmaf(w[c1 * 3 + 0], a, s0);
                s1 = fmaf(w[c1 * 3 + 1], a, s1);
                s2 = fmaf(w[c1 * 3 + 2], a, s2);
            }
            float bb = b2[ic * 32 + c2];
            v_int_s[c2] = leaky(bb + s0 + s1 + s2);  // h = 1..21
            v_top_s[c2] = leaky(bb + s1 + s2);       // h = 0
            v_bot_s[c2] = leaky(bb + s0 + s1);       // h = 22
        }
        __syncthreads();

        // conv3 tap sums over c2; lane handles c3 = lane and lane+32.
        // w=1 output rows: h=0 (taps kh=1->top, kh=2->int), h=1..10 (all int),
        // h=11 (kh=0->int, kh=1->bot). w=0 column is pure leaky(b3).
        for (int rep = 0; rep < 2; ++rep) {
            int c3 = lane + rep * 32;
            const float* w = W3 + ((size_t)ic * 64 + c3) * 32 * 3;
            float t0i = 0.f, t1i = 0.f, t2i = 0.f, t1t = 0.f, t1b = 0.f;
            #pragma unroll 8
            for (int c2 = 0; c2 < 32; ++c2) {
                float w0 = w[c2 * 3 + 0];
                float w1 = w[c2 * 3 + 1];
                float w2v = w[c2 * 3 + 2];
                float vi = v_int_s[c2];
                t0i = fmaf(w0,  vi,          t0i);
                t1i = fmaf(w1,  vi,          t1i);
                t2i = fmaf(w2v, vi,          t2i);
                t1t = fmaf(w1,  v_top_s[c2], t1t);
                t1b = fmaf(w1,  v_bot_s[c2], t1b);
            }
            float b3v  = b3[ic * 64 + c3];
            float oh0  = leaky(b3v + t1t + t2i);
            float oin  = leaky(b3v + t0i + t1i + t2i);
            float oh11 = leaky(b3v + t0i + t1b);
            float ow0  = leaky(b3v);
            // global mean over 12x2 outputs.
            mean_s[e][c3] = (12.0f * ow0 + oh0 + 10.0f * oin + oh11) * (1.0f / 24.0f);
        }
        __syncthreads();
    }

    // FC for 16 experts at once via f32 WMMA (16x16x4, K=64 -> 16 steps).
    // A[m,k] = Wfc[i0+m, k], B[k,n] = mean_s[n][k]; diag(D) = FC pre-acts.
    // A frag (ISA 7.12.2): lane half selects K pair {0,1}/{2,3}; m = lane&15.
    // B frag mirrored: lane&15 = column n, lane half selects K pair.
    const int half = lane >> 4;
    const int m    = lane & 15;
    const int koff = half * 2;
    int ia = i0 + m; if (ia >= N_MODELS) ia = N_MODELS - 1;
    const float* wrow = Wfc + (size_t)ia * 64;

    v8f acc = {0.f, 0.f, 0.f, 0.f, 0.f, 0.f, 0.f, 0.f};
    #pragma unroll
    for (int s = 0; s < 16; ++s) {
        int k = 4 * s + koff;
        v2f a; a.x = wrow[k];        a.y = wrow[k + 1];
        v2f b; b.x = mean_s[m][k];   b.y = mean_s[m][k + 1];
        acc = __builtin_amdgcn_wmma_f32_16x16x4_f32(
            /*neg_a=*/false, a, /*neg_b=*/false, b,
            /*c_mod=*/(short)0, acc, /*reuse_a=*/false, /*reuse_b=*/false);
    }
    // D layout: VGPR r -> M = r + 8*half, N = lane&15.
    #pragma unroll
    for (int r = 0; r < 8; ++r)
        dmat[(r + 8 * half) * 16 + m] = acc[r];
    __syncthreads();

    if (lane < 16) {
        int i = i0 + lane;
        if (i < N_MODELS)
            consts[i] = leaky(dmat[lane * 16 + lane] + bfc[i]);
    }
}

// Broadcast: out[b,i] = consts[i]. lcm(90,4)=180, so two rows = 45 distinct
// float4 patterns; build them once in LDS, then stream b128 stores.
__global__ void __launch_bounds__(256)
broadcast_kernel(const float* __restrict__ consts,
                 float4* __restrict__ out4, int n4) {
    __shared__ float4 lds4[45];
    int t = threadIdx.x;
    if (t < 45) {
        int q = 4 * t;
        float4 v;
        v.x = consts[(q + 0) % 90];
        v.y = consts[(q + 1) % 90];
        v.z = consts[(q + 2) % 90];
        v.w = consts[(q + 3) % 90];
        lds4[t] = v;
    }
    __syncthreads();
    int idx = blockIdx.x * 256 + t;
    if (idx < n4)
        out4[idx] = lds4[idx % 45];
}

extern "C" void kernel_launch(void* const* d_in, const int* in_sizes, int n_in,
                              void* d_out, int out_size, void* d_ws, size_t ws_size,
                              hipStream_t stream) {
    (void)in_sizes; (void)n_in; (void)ws_size;
    // setup_inputs order: x, W1, b1, W2, b2, W3, b3, Wfc, bfc.
    // x and W1 are mathematically dead (stride-2 convs only sample width padding).
    const float* b1  = (const float*)d_in[2];
    const float* W2  = (const float*)d_in[3];
    const float* b2  = (const float*)d_in[4];
    const float* W3  = (const float*)d_in[5];
    const float* b3  = (const float*)d_in[6];
    const float* Wfc = (const float*)d_in[7];
    const float* bfc = (const float*)d_in[8];

    float* consts = (float*)d_ws;  // 90 floats of scratch

    // 6 blocks x 16 experts = 96 slots (last 6 masked).
    expert_const_kernel<<<6, 32, 0, stream>>>(b1, W2, b2, W3, b3, Wfc, bfc, consts);

    int n4 = out_size / 4;  // 2048*90/4 = 46080 float4 stores (737 KB)
    broadcast_kernel<<<(n4 + 255) / 256, 256, 0, stream>>>(consts, (float4*)d_out, n4);
}